// CustomProposalLayer_70866960384558
// MI455X (gfx1250) — compile-verified
//
#include <hip/hip_runtime.h>
#include <hip/hip_bf16.h>

#define DEV_INLINE __device__ __forceinline__

// ---------------- problem constants ----------------
constexpr int B  = 32;
constexpr int A  = 4;
constexpr int W2 = 152, W3 = 76, W4 = 38, W5 = 19;
constexpr int C2 = W2 * W2;   // 23104
constexpr int C3 = W3 * W3;   // 5776
constexpr int C4 = W4 * W4;   // 1444
constexpr int C5 = W5 * W5;   // 361
constexpr int N2 = A * C2;    // 92416
constexpr int N3 = A * C3;    // 23104
constexpr int N4 = A * C4;    // 5776
constexpr int N5 = A * C5;    // 1444
constexpr int P  = N2 + N3 + N4 + N5;   // 122740 proposals per batch
constexpr int MAX_PROP = 100;
constexpr float SCORE_THRESH = 0.25f;
constexpr float IOU_THRESH   = 0.5f;

__constant__ float c_anchors[4][4][2] = {
    {{12, 16}, {19, 36}, {40, 28}, {36, 75}},
    {{36, 75}, {76, 55}, {72, 146}, {142, 110}},
    {{72, 146}, {142, 110}, {192, 243}, {459, 401}},
    {{142, 110}, {192, 243}, {300, 300}, {459, 401}},
};

// ---------------- CDNA5 async-to-LDS helpers ----------------
// Round-1 diagnostic proved the gfx1250 device pass has
// __builtin_amdgcn_global_load_async_to_lds_b128 with signature
// (int4 AS1* gaddr, int4 AS3* lds, imm offset, imm cpol).
// __has_builtin resolves per compilation pass: true on the gfx1250 device
// pass (async path taken), false on the x86 host pass (parse-only fallback).
typedef int v4i __attribute__((__vector_size__(16)));
typedef __attribute__((address_space(1))) v4i* gptr_v4i;
typedef __attribute__((address_space(3))) v4i* lptr_v4i;

// Per-lane DMA of 16B from global to LDS, tracked by ASYNCcnt (gfx1250 path).
DEV_INLINE void async_copy16(const float4* gsrc, float4* ldst) {
#if __has_builtin(__builtin_amdgcn_global_load_async_to_lds_b128)
    __builtin_amdgcn_global_load_async_to_lds_b128(
        (gptr_v4i)(const_cast<float4*>(gsrc)),
        (lptr_v4i)(ldst),
        /*offset=*/0, /*cpol=*/0);
#else
    // host-pass parse-only fallback (never executed on device)
    *ldst = *gsrc;
#endif
}

DEV_INLINE void wait_async0() {
#if __has_builtin(__builtin_amdgcn_s_wait_asynccnt)
    __builtin_amdgcn_s_wait_asynccnt(0);
#endif
}

// ---------------- decode kernel ----------------
// One thread per proposal: sigmoid/exp decode into [x1,y1,x2,y2], thresholded score.
template <int Wd>
DEV_INLINE void decode_one(const float* __restrict__ src, int b, int li,
                           float stride, int lvl, float4& box, float& score) {
    constexpr int cells = Wd * Wd;
    int a   = li / cells;                 // compile-time divisor -> magic numbers
    int rem = li - a * cells;
    int y   = rem / Wd;
    int x   = rem - y * Wd;
    const float* q = src + ((size_t)(b * A + a) * cells + rem) * 6;
    float tx = q[0], ty = q[1], tw = q[2], th = q[3], cf = q[4], cl = q[5];
    float sx = 1.f / (1.f + __expf(-tx));
    float sy = 1.f / (1.f + __expf(-ty));
    float cx = (sx + (float)x) * stride;
    float cy = (sy + (float)y) * stride;
    float w  = __expf(tw) * c_anchors[lvl][a][0];
    float h  = __expf(th) * c_anchors[lvl][a][1];
    float s  = (1.f / (1.f + __expf(-cf))) * (1.f / (1.f + __expf(-cl)));
    box   = make_float4(cx - 0.5f * w, cy - 0.5f * h, cx + 0.5f * w, cy + 0.5f * h);
    score = s;
}

// 2D grid: blockIdx.y = batch, blockIdx.x tiles P. Keeps all indexing 32-bit
// (no 64-bit magic-division in the prologue).
__global__ __launch_bounds__(256) void decode_kernel(
    const float* __restrict__ p2, const float* __restrict__ p3,
    const float* __restrict__ p4, const float* __restrict__ p5,
    float4* __restrict__ boxes, float* __restrict__ scores) {
    const int i = blockIdx.x * 256 + threadIdx.x;
    const int b = blockIdx.y;
    if (i >= P) return;

    float4 box;
    float  s;
    if (i < N2)                  decode_one<W2>(p2, b, i,                 4.f, 0, box, s);
    else if (i < N2 + N3)        decode_one<W3>(p3, b, i - N2,            8.f, 1, box, s);
    else if (i < N2 + N3 + N4)   decode_one<W4>(p4, b, i - (N2 + N3),    16.f, 2, box, s);
    else                         decode_one<W5>(p5, b, i - (N2 + N3 + N4), 32.f, 3, box, s);

    size_t o  = (size_t)b * P + i;
    boxes[o]  = box;
    scores[o] = (s >= SCORE_THRESH) ? s : 0.f;   // s0 = where(score>=thresh, score, 0)
}

// ---------------- fused NMS kernel ----------------
// One 512-thread block (16 wave32s) per batch. 100 rounds of
// fused {suppress with round-k box} + {argmax for round k+1}.
// Boxes streamed through double-buffered LDS tiles via async-to-LDS DMA;
// the prefetch of tile t+1 overlaps IoU compute on tile t.
constexpr int NMS_THREADS = 512;
constexpr int TILE        = 1024;                       // proposals per LDS tile (16KB)
constexpr int NTILES      = (P + TILE - 1) / TILE;      // 120
constexpr int VPT         = TILE / NMS_THREADS;         // 2 float4 per thread per tile

DEV_INLINE void prefetch_tile(const float4* __restrict__ gbase, int tstart,
                              float4* buf, int tid) {
#pragma unroll
    for (int u = 0; u < VPT; ++u) {
        int j = tstart + tid + u * NMS_THREADS;
        if (j < P) async_copy16(gbase + j, &buf[tid + u * NMS_THREADS]);
    }
}

__global__ __launch_bounds__(NMS_THREADS) void nms_kernel(
    const float4* __restrict__ boxes, float* __restrict__ scores,
    float* __restrict__ out) {
    __shared__ float4 tilebuf[2][TILE];       // 32 KB double buffer
    __shared__ float  redV[NMS_THREADS];
    __shared__ int    redI[NMS_THREADS];
    __shared__ float  selBox[4];
    __shared__ float  selScore;

    const int b   = blockIdx.x;
    const int tid = threadIdx.x;
    const float4* __restrict__ bbx = boxes + (size_t)b * P;
    float* __restrict__ sc         = scores + (size_t)b * P;
    float* __restrict__ orow       = out + (size_t)b * MAX_PROP * 5;

    // ---- initial argmax (round 0 selection) ----
    float bestV = -1.f;
    int   bestI = 0;
    for (int j = tid; j < P; j += NMS_THREADS) {
        float s = sc[j];
        if (s > bestV) { bestV = s; bestI = j; }   // strict > keeps earliest per thread
    }
    redV[tid] = bestV; redI[tid] = bestI;
    __syncthreads();
    for (int off = NMS_THREADS / 2; off > 0; off >>= 1) {
        if (tid < off) {
            float v2 = redV[tid + off]; int i2 = redI[tid + off];
            if (v2 > redV[tid] || (v2 == redV[tid] && i2 < redI[tid])) {
                redV[tid] = v2; redI[tid] = i2;
            }
        }
        __syncthreads();
    }
    if (tid == 0) {
        float4 bb = bbx[redI[0]];
        selBox[0] = bb.x; selBox[1] = bb.y; selBox[2] = bb.z; selBox[3] = bb.w;
        selScore  = redV[0];
    }
    __syncthreads();

    // ---- 100 fused rounds ----
    for (int k = 0; k < MAX_PROP; ++k) {
        const float sv  = selScore;
        const float bx1 = selBox[0], by1 = selBox[1], bx2 = selBox[2], by2 = selBox[3];
        const bool  keep = sv > 0.f;          // block-uniform (from shared)

        if (tid == 0) {
            float* r = orow + k * 5;
            r[0] = keep ? bx1 : 0.f;
            r[1] = keep ? by1 : 0.f;
            r[2] = keep ? bx2 : 0.f;
            r[3] = keep ? by2 : 0.f;
            r[4] = keep ? sv  : 0.f;
        }

        bestV = -1.f; bestI = 0;
        if (keep) {
            const float a1 = fmaxf(bx2 - bx1, 0.f) * fmaxf(by2 - by1, 0.f);
            prefetch_tile(bbx, 0, tilebuf[0], tid);          // kick tile 0
            for (int t = 0; t < NTILES; ++t) {
                wait_async0();                                // my copies of tile t done
                __syncthreads();                              // everyone's copies visible
                if (t + 1 < NTILES)                           // overlap next DMA w/ compute
                    prefetch_tile(bbx, (t + 1) * TILE, tilebuf[(t + 1) & 1], tid);

                const float4* tb   = tilebuf[t & 1];
                const int     base = t * TILE;
#pragma unroll
                for (int u = 0; u < VPT; ++u) {
                    int j = base + tid + u * NMS_THREADS;
                    if (j < P) {
                        float s = sc[j];
                        if (s > 0.f) {
                            float4 c   = tb[tid + u * NMS_THREADS];
                            float ix1  = fmaxf(bx1, c.x), iy1 = fmaxf(by1, c.y);
                            float ix2  = fminf(bx2, c.z), iy2 = fminf(by2, c.w);
                            float iw   = fmaxf(ix2 - ix1, 0.f);
                            float ih   = fmaxf(iy2 - iy1, 0.f);
                            float inter = iw * ih;
                            float a2   = fmaxf(c.z - c.x, 0.f) * fmaxf(c.w - c.y, 0.f);
                            float iou  = inter / (a1 + a2 - inter + 1e-9f);
                            if (iou > IOU_THRESH) { s = 0.f; sc[j] = 0.f; }  // suppress
                        }
                        if (s > bestV) { bestV = s; bestI = j; }             // next argmax
                    }
                }
            }
            __syncthreads();   // all waves done with buffers before next round reuses them
        } else {
            // all remaining scores are zero; argmax of all-zero array is index 0
            bestV = 0.f; bestI = 0;
        }

        // ---- block reduction -> selection for round k+1 ----
        redV[tid] = bestV; redI[tid] = bestI;
        __syncthreads();
        for (int off = NMS_THREADS / 2; off > 0; off >>= 1) {
            if (tid < off) {
                float v2 = redV[tid + off]; int i2 = redI[tid + off];
                if (v2 > redV[tid] || (v2 == redV[tid] && i2 < redI[tid])) {
                    redV[tid] = v2; redI[tid] = i2;
                }
            }
            __syncthreads();
        }
        if (tid == 0) {
            float4 bb = bbx[redI[0]];
            selBox[0] = bb.x; selBox[1] = bb.y; selBox[2] = bb.z; selBox[3] = bb.w;
            selScore  = fmaxf(redV[0], 0.f);
        }
        __syncthreads();
    }
}

// ---------------- launcher ----------------
extern "C" void kernel_launch(void* const* d_in, const int* in_sizes, int n_in,
                              void* d_out, int out_size, void* d_ws, size_t ws_size,
                              hipStream_t stream) {
    const float* p2 = (const float*)d_in[0];
    const float* p3 = (const float*)d_in[1];
    const float* p4 = (const float*)d_in[2];
    const float* p5 = (const float*)d_in[3];

    // workspace layout: [boxes float4[B*P] | scores float[B*P]]  (~78.5 MB, L2-resident)
    float4* boxes  = (float4*)d_ws;
    float*  scores = (float*)((char*)d_ws + (size_t)B * P * sizeof(float4));

    dim3 dgrid((P + 255) / 256, B);
    decode_kernel<<<dgrid, 256, 0, stream>>>(p2, p3, p4, p5, boxes, scores);

    nms_kernel<<<B, NMS_THREADS, 0, stream>>>(boxes, scores, (float*)d_out);
}